// GCN_47880295415876
// MI455X (gfx1250) — compile-verified
//
#include <hip/hip_runtime.h>
#include <hip/hip_bf16.h>

typedef __attribute__((ext_vector_type(2))) float v2f;
typedef __attribute__((ext_vector_type(8))) float v8f;

#define HDIM 64
#define WSTRIDE 68   // padded LDS row stride (floats): conflict-free b64 reads

// ---------------------------------------------------------------------------
// init: deg[n]=1 (self loop), zero graph-pool accumulators
// ---------------------------------------------------------------------------
__global__ void k_init(float* __restrict__ deg, float* __restrict__ gsum,
                       float* __restrict__ gcnt, int n, int g) {
    int i = blockIdx.x * blockDim.x + threadIdx.x;
    if (i < n) deg[i] = 1.0f;
    if (i < g * HDIM) gsum[i] = 0.0f;
    if (i < g) gcnt[i] = 0.0f;
}

// deg[dst] += 1 per edge
__global__ void k_deg(const int* __restrict__ dst, float* __restrict__ deg, int e) {
    int i = blockIdx.x * blockDim.x + threadIdx.x;
    if (i < e) atomicAdd(&deg[dst[i]], 1.0f);
}

__global__ void k_dinv(const float* __restrict__ deg, float* __restrict__ dinv, int n) {
    int i = blockIdx.x * blockDim.x + threadIdx.x;
    if (i < n) dinv[i] = rsqrtf(deg[i]);   // deg >= 1 always (self loop)
}

// ---------------------------------------------------------------------------
// embedding: h[n][c] = x[n][0:4] . W_emb[0:4][c] + b_emb[c]
// ---------------------------------------------------------------------------
__global__ void k_embed(const float* __restrict__ x, const float* __restrict__ W,
                        const float* __restrict__ b, float* __restrict__ h, int n) {
    int i = blockIdx.x * blockDim.x + threadIdx.x;
    if (i >= n * HDIM) return;
    int node = i >> 6, c = i & 63;
    float4 xv = ((const float4*)x)[node];
    h[i] = b[c] + xv.x * W[c] + xv.y * W[HDIM + c]
                + xv.z * W[2 * HDIM + c] + xv.w * W[3 * HDIM + c];
}

// ---------------------------------------------------------------------------
// WMMA f32 GEMM: out[rows,64] = in[rows,64] @ W[64,64] (+bias, relu optional)
// One wave = 16-row x 64-col tile; K swept with V_WMMA_F32_16X16X4_F32.
// W is stored TRANSPOSED in LDS (sWt[n][k], stride 68) so each B fragment
// {W[ka][n], W[ka+1][n]} is one contiguous bank-conflict-free ds_load_b64.
// ---------------------------------------------------------------------------
__global__ void k_gemm_wmma(const float* __restrict__ in, const float* __restrict__ W,
                            const float* __restrict__ bias, float* __restrict__ out,
                            int rows, int relu) {
    __shared__ float sWt[HDIM * WSTRIDE];
    // cooperative transpose-load: thread reads W[k][n], writes sWt[n*68 + k]
    for (int i = threadIdx.x; i < HDIM * HDIM; i += blockDim.x) {
        int k = i >> 6, ncol = i & 63;
        sWt[ncol * WSTRIDE + k] = W[i];
    }
    __syncthreads();

    const int wave = threadIdx.x >> 5;
    const int lane = threadIdx.x & 31;
    const int row0 = (blockIdx.x * 8 + wave) * 16;
    if (row0 >= rows) return;

    const int hsel = lane >> 4;                // 0: K+{0,1}  1: K+{2,3}
    const int l16  = lane & 15;

    v8f acc[4];
    acc[0] = {}; acc[1] = {}; acc[2] = {}; acc[3] = {};

    const float* arow = in + (size_t)(row0 + l16) * HDIM;
#pragma unroll
    for (int kk = 0; kk < HDIM; kk += 4) {
        const int ka = kk + hsel * 2;
        v2f a = *(const v2f*)&arow[ka];        // A 16x4 fragment (ISA layout)
#pragma unroll
        for (int t = 0; t < 4; ++t) {
            const int ncol = t * 16 + l16;
            v2f b = *(const v2f*)&sWt[ncol * WSTRIDE + ka];   // B 4x16 fragment
            acc[t] = __builtin_amdgcn_wmma_f32_16x16x4_f32(
                false, a, false, b, (short)0, acc[t], false, false);
        }
    }

    // C/D layout: VGPR v -> row v + 8*hsel, col = l16 + 16*t
#pragma unroll
    for (int t = 0; t < 4; ++t) {
        const int ncol = t * 16 + l16;
        float badd = bias ? bias[ncol] : 0.0f;
#pragma unroll
        for (int v = 0; v < 8; ++v) {
            float val = acc[t][v] + badd;
            if (relu) val = fmaxf(val, 0.0f);
            out[(size_t)(row0 + v + 8 * hsel) * HDIM + ncol] = val;
        }
    }
}

// ---------------------------------------------------------------------------
// self-loop: hout[n][c] = hw[n][c] * dinv[n]^2   (dense init, no atomics)
// ---------------------------------------------------------------------------
__global__ void k_selfloop(const float* __restrict__ hw, const float* __restrict__ dinv,
                           float* __restrict__ hout, int n) {
    int i = blockIdx.x * blockDim.x + threadIdx.x;
    if (i >= n * HDIM) return;
    float di = dinv[i >> 6];
    hout[i] = hw[i] * di * di;
}

// scatter: one wave32 per edge; lane c handles channels {2c, 2c+1} (b64 gather)
__global__ void k_scatter(const float* __restrict__ hw, const int* __restrict__ src,
                          const int* __restrict__ dst, const float* __restrict__ dinv,
                          float* __restrict__ hout, int e) {
    int edge = blockIdx.x * (blockDim.x >> 5) + (threadIdx.x >> 5);
    if (edge >= e) return;
    int lane = threadIdx.x & 31;
    int s = src[edge], d = dst[edge];
    float w = dinv[s] * dinv[d];
    const float2 hv = ((const float2*)(hw + (size_t)s * HDIM))[lane];
    float* hd = hout + (size_t)d * HDIM + 2 * lane;
    atomicAdd(&hd[0], hv.x * w);
    atomicAdd(&hd[1], hv.y * w);
}

__global__ void k_bias_relu(float* __restrict__ h, const float* __restrict__ b, int n) {
    int i = blockIdx.x * blockDim.x + threadIdx.x;
    if (i >= n * HDIM) return;
    h[i] = fmaxf(h[i] + b[i & 63], 0.0f);
}

// ---------------------------------------------------------------------------
// global mean pool
// ---------------------------------------------------------------------------
__global__ void k_pool(const float* __restrict__ h, const int* __restrict__ batch,
                       float* __restrict__ gsum, float* __restrict__ gcnt, int n) {
    int i = blockIdx.x * blockDim.x + threadIdx.x;
    if (i >= n * HDIM) return;
    int node = i >> 6, c = i & 63;
    int g = batch[node];
    atomicAdd(&gsum[g * HDIM + c], h[i]);
    if (c == 0) atomicAdd(&gcnt[g], 1.0f);
}

__global__ void k_graphx(const float* __restrict__ gsum, const float* __restrict__ gcnt,
                         float* __restrict__ gx, int g) {
    int i = blockIdx.x * blockDim.x + threadIdx.x;
    if (i >= g * HDIM) return;
    gx[i] = gsum[i] / fmaxf(gcnt[i >> 6], 1.0f);
}

// pred[g,3] = gh[g,64] @ W_l2[64,3] + b_l2
__global__ void k_final(const float* __restrict__ gh, const float* __restrict__ W,
                        const float* __restrict__ b, float* __restrict__ out, int g) {
    int i = blockIdx.x * blockDim.x + threadIdx.x;
    if (i >= g * 3) return;
    int gi = i / 3, j = i - gi * 3;
    const float* r = gh + (size_t)gi * HDIM;
    float acc = b[j];
#pragma unroll
    for (int c = 0; c < HDIM; ++c) acc += r[c] * W[c * 3 + j];
    out[i] = acc;
}

// ---------------------------------------------------------------------------
extern "C" void kernel_launch(void* const* d_in, const int* in_sizes, int n_in,
                              void* d_out, int out_size, void* d_ws, size_t ws_size,
                              hipStream_t stream) {
    const float* x      = (const float*)d_in[0];
    const int*   edge   = (const int*)  d_in[1];   // [2,E]: row0=src, row1=dst
    const int*   batch  = (const int*)  d_in[2];
    const float* W_emb  = (const float*)d_in[3];
    const float* b_emb  = (const float*)d_in[4];
    const float* W_conv = (const float*)d_in[5];   // [L,64,64]
    const float* b_conv = (const float*)d_in[6];   // [L,64]
    const float* W_l1   = (const float*)d_in[7];
    const float* b_l1   = (const float*)d_in[8];
    const float* W_l2   = (const float*)d_in[9];
    const float* b_l2   = (const float*)d_in[10];
    float* out = (float*)d_out;

    const int N = in_sizes[0] / 4;
    const int E = in_sizes[1] / 2;
    const int L = in_sizes[5] / (HDIM * HDIM);
    const int G = out_size / 3;

    const int* srcp = edge;
    const int* dstp = edge + E;

    // workspace carve (floats)
    float* ws   = (float*)d_ws;
    float* deg  = ws;
    float* dinv = deg  + N;
    float* bufA = dinv + N;                       // h (current)
    float* bufB = bufA + (size_t)N * HDIM;        // hw (gemm out)
    float* gsum = bufB + (size_t)N * HDIM;
    float* gcnt = gsum + (size_t)G * HDIM;
    float* gx   = gcnt + G;
    float* gh   = gx   + (size_t)G * HDIM;

    const int T = 256;
    auto cdiv = [](int a, int b) { return (a + b - 1) / b; };
    const int NH = N * HDIM;

    // degrees & symmetric normalization
    {
        int cover = (N > G * HDIM) ? N : G * HDIM;
        k_init<<<cdiv(cover, T), T, 0, stream>>>(deg, gsum, gcnt, N, G);
    }
    k_deg <<<cdiv(E, T), T, 0, stream>>>(dstp, deg, E);
    k_dinv<<<cdiv(N, T), T, 0, stream>>>(deg, dinv, N);

    // node embedding
    k_embed<<<cdiv(NH, T), T, 0, stream>>>(x, W_emb, b_emb, bufA, N);

    // GCN layers
    const int gemm_blocks_N = cdiv(N, 128);          // 8 waves/block, 16 rows/wave
    for (int l = 0; l < L; ++l) {
        k_gemm_wmma<<<gemm_blocks_N, T, 0, stream>>>(
            bufA, W_conv + (size_t)l * HDIM * HDIM, nullptr, bufB, N, 0);
        k_selfloop<<<cdiv(NH, T), T, 0, stream>>>(bufB, dinv, bufA, N);
        k_scatter <<<cdiv(E, T / 32), T, 0, stream>>>(bufB, srcp, dstp, dinv, bufA, E);
        k_bias_relu<<<cdiv(NH, T), T, 0, stream>>>(bufA, b_conv + (size_t)l * HDIM, N);
    }

    // pooling + MLP head
    k_pool  <<<cdiv(NH, T), T, 0, stream>>>(bufA, batch, gsum, gcnt, N);
    k_graphx<<<cdiv(G * HDIM, T), T, 0, stream>>>(gsum, gcnt, gx, G);
    k_gemm_wmma<<<cdiv(G, 128), T, 0, stream>>>(gx, W_l1, b_l1, gh, G, 1);
    k_final <<<cdiv(G * 3, T), T, 0, stream>>>(gh, W_l2, b_l2, out, G);
}